// ProductionKasminaLayer_40261023433118
// MI455X (gfx1250) — compile-verified
//
#include <hip/hip_runtime.h>
#include <hip/hip_bf16.h>
#include <math.h>

#define BDIM 512
#define HDIM 8192
#define SDIM 64
#define CDIM 128
#define SPLITB 16
#define ROWS_PER_BLOCK (BDIM / SPLITB)          // 32
#define NWAVES 8
#define ROWS_PER_WAVE (ROWS_PER_BLOCK / NWAVES) // 4
#define NBUF 4                                  // async pipeline depth
#define MAX_OFF ((SDIM - 1) * CDIM)             // 8064

__device__ __forceinline__ float wave_sum32(float v) {
#pragma unroll
  for (int m = 16; m > 0; m >>= 1) v += __shfl_xor(v, m, 32);
  return v;
}

template <int N>
__device__ __forceinline__ void wait_async() {
  asm volatile("s_wait_asynccnt %0" :: "i"(N) : "memory");
}

__global__ void kasmina_zero_tail(float* __restrict__ p, int n) {
  int i = blockIdx.x * blockDim.x + threadIdx.x;
  if (i < n) p[i] = 0.0f;
}

__global__ __launch_bounds__(256) void kasmina_bwd_main(
    const float* __restrict__ g, const float* __restrict__ x,
    const float* __restrict__ alpha, const float* __restrict__ bwts,
    const float* __restrict__ bscales, const int* __restrict__ lcs,
    const int* __restrict__ ids, const int* __restrict__ bts,
    const int* __restrict__ offsets, float* __restrict__ out_gi,
    float* __restrict__ out_gbw, float* __restrict__ out_gbs) {
  __shared__ __align__(16) float lds_g[NBUF][NWAVES][CDIM];
  __shared__ __align__(16) float lds_x[NBUF][NWAVES][CDIM];
  __shared__ __align__(16) float s_bw[CDIM];
  __shared__ __align__(16) float s_aw[CDIM];
  __shared__ float accT[CDIM], accTm[CDIM], accG[CDIM], accGm[CDIM];
  __shared__ float accP[2];
  __shared__ float red[CDIM];
  __shared__ float s_scal[4];

  const int tid = threadIdx.x;
  const int s = blockIdx.x / SPLITB;     // segment
  const int chunk = blockIdx.x % SPLITB; // batch chunk

  // ---- per-segment metadata (uniform across block) ----
  const int bid = ids[s];
  const int bidc = min(max(bid, 0), SDIM - 1);
  const int off = offsets[bidc];
  const float a = alpha[s];
  const float bscale = bscales[bidc];
  const int bt = bts[s];
  const int lsv = lcs[s];
  const bool act = (lsv >= 3) && (bid > 0) && (bid < SDIM) && (off <= MAX_OFF);

  // ---- blueprint row + softmax (128 lanes), zero accumulators ----
  float v = 0.f;
  if (tid < CDIM) {
    v = bwts[off + tid];
    s_bw[tid] = v;
    red[tid] = v;
    accT[tid] = 0.f; accTm[tid] = 0.f; accG[tid] = 0.f; accGm[tid] = 0.f;
  }
  if (tid < 2) accP[tid] = 0.f;
  __syncthreads();
  for (int st = 64; st > 0; st >>= 1) {
    if (tid < st) red[tid] = fmaxf(red[tid], red[tid + st]);
    __syncthreads();
  }
  const float mx = red[0];
  __syncthreads();
  float e = 0.f;
  if (tid < CDIM) { e = __expf(v - mx); red[tid] = e; }
  __syncthreads();
  for (int st = 64; st > 0; st >>= 1) {
    if (tid < st) red[tid] += red[tid + st];
    __syncthreads();
  }
  const float esum = red[0];
  if (tid < CDIM) s_aw[tid] = e / esum;
  __syncthreads();

  const int wv = tid >> 5, lane = tid & 31, c0 = lane << 2;
  const float4 bw4 = *(const float4*)&s_bw[c0];
  const float4 aw4 = *(const float4*)&s_aw[c0];
  const float w0 = s_bw[0], w1 = s_bw[1], w2 = s_bw[2];
  const float oma = 1.0f - a;
  const int bbase = chunk * ROWS_PER_BLOCK;
  const unsigned rowbyte = (unsigned)(s * CDIM + c0) * 4u;

  const unsigned long long gp = (unsigned long long)(size_t)g;
  const unsigned long long xp = (unsigned long long)(size_t)x;

  float T0=0,T1=0,T2=0,T3=0, Tm0=0,Tm1=0,Tm2=0,Tm3=0;
  float G0=0,G1=0,G2=0,G3=0, Gm0=0,Gm1=0,Gm2=0,Gm3=0;
  float p1=0.f, p2=0.f;

  // ---- 4-deep async global->LDS pipeline (CDNA5 ASYNCcnt path) ----
  auto issue = [&](int i) {
    int b = bbase + i * NWAVES + wv;
    unsigned voff = (unsigned)b * (HDIM * 4u) + rowbyte;
    unsigned lg = (unsigned)(size_t)&lds_g[i & (NBUF - 1)][wv][c0];
    unsigned lx = (unsigned)(size_t)&lds_x[i & (NBUF - 1)][wv][c0];
    asm volatile("global_load_async_to_lds_b128 %0, %1, %2"
                 :: "v"(lg), "v"(voff), "s"(gp) : "memory");
    asm volatile("global_load_async_to_lds_b128 %0, %1, %2"
                 :: "v"(lx), "v"(voff), "s"(xp) : "memory");
  };

  auto consume = [&](int i) {
    const int buf = i & (NBUF - 1);
    const int b = bbase + i * NWAVES + wv;
    const float* gr = &lds_g[buf][wv][0];
    const float* xr = &lds_x[buf][wv][0];
    float4 g4 = *(const float4*)(gr + c0);
    float4 x4 = *(const float4*)(xr + c0);
    float gnp1 = (lane < 31) ? gr[c0 + 4] : 0.f;
    float gnp2 = (lane < 31) ? gr[c0 + 5] : 0.f;
    float xm1  = (lane > 0)  ? xr[c0 - 1] : 0.f;
    float xm2  = (lane > 0)  ? xr[c0 - 2] : 0.f;

    // grad clip: per-row L2 norm via wave32 cross-lane reduction
    float n2 = g4.x*g4.x + g4.y*g4.y + g4.z*g4.z + g4.w*g4.w;
    n2 = wave_sum32(n2);
    float clipf = fminf(1.0f / (sqrtf(n2) + 1e-6f), 1.0f);
    float gcx = g4.x*clipf, gcy = g4.y*clipf, gcz = g4.z*clipf, gcw = g4.w*clipf;

    float rmx = (x4.x * bw4.x > 0.f) ? 1.f : 0.f;
    float rmy = (x4.y * bw4.y > 0.f) ? 1.f : 0.f;
    float rmz = (x4.z * bw4.z > 0.f) ? 1.f : 0.f;
    float rmw = (x4.w * bw4.w > 0.f) ? 1.f : 0.f;

    float gix, giy, giz, giw;
    if (!act) {
      gix = gcx; giy = gcy; giz = gcz; giw = gcw;
    } else if (bt == 0) {
      float f = oma + a * bscale;
      gix = gcx*f; giy = gcy*f; giz = gcz*f; giw = gcw*f;
    } else if (bt == 1) {
      gix = gcx*(oma + a*aw4.x); giy = gcy*(oma + a*aw4.y);
      giz = gcz*(oma + a*aw4.z); giw = gcw*(oma + a*aw4.w);
    } else if (bt == 2) {
      float ab = a * bscale;
      gix = gcx*(oma + ab*bw4.x*rmx); giy = gcy*(oma + ab*bw4.y*rmy);
      giz = gcz*(oma + ab*bw4.z*rmz); giw = gcw*(oma + ab*bw4.w*rmw);
    } else {
      float ab = a * bscale;  // conv taps use raw g
      gix = oma*gcx + ab*(w0*gcx + w1*g4.y + w2*g4.z);
      giy = oma*gcy + ab*(w0*gcy + w1*g4.z + w2*g4.w);
      giz = oma*gcz + ab*(w0*gcz + w1*g4.w + w2*gnp1);
      giw = oma*gcw + ab*(w0*gcw + w1*gnp1 + w2*gnp2);
    }
    float4 giv; giv.x = gix; giv.y = giy; giv.z = giz; giv.w = giw;
    *(float4*)(out_gi + (size_t)b * HDIM + (size_t)s * CDIM + c0) = giv;

    // batch-partial accumulators (all outputs are linear in these)
    T0 += gcx*x4.x; T1 += gcy*x4.y; T2 += gcz*x4.z; T3 += gcw*x4.w;
    G0 += gcx; G1 += gcy; G2 += gcz; G3 += gcw;
    Tm0 += gcx*x4.x*rmx; Tm1 += gcy*x4.y*rmy; Tm2 += gcz*x4.z*rmz; Tm3 += gcw*x4.w*rmw;
    Gm0 += gcx*rmx; Gm1 += gcy*rmy; Gm2 += gcz*rmz; Gm3 += gcw*rmw;
    p1 += gcx*xm1 + gcy*x4.x + gcz*x4.y + gcw*x4.z;
    p2 += gcx*xm2 + gcy*xm1 + gcz*x4.x + gcw*x4.y;
  };

  // hand-sequenced 4-stage pipeline: prefetch distance 3, 2 loads per row
  issue(0); issue(1); issue(2);
  issue(3); wait_async<6>(); consume(0);
  wait_async<4>(); consume(1);
  wait_async<2>(); consume(2);
  wait_async<0>(); consume(3);

  // ---- cross-wave reduction via LDS f32 atomics (ds_add_f32) ----
  atomicAdd(&accT[c0+0], T0);  atomicAdd(&accT[c0+1], T1);
  atomicAdd(&accT[c0+2], T2);  atomicAdd(&accT[c0+3], T3);
  atomicAdd(&accTm[c0+0], Tm0); atomicAdd(&accTm[c0+1], Tm1);
  atomicAdd(&accTm[c0+2], Tm2); atomicAdd(&accTm[c0+3], Tm3);
  atomicAdd(&accG[c0+0], G0);  atomicAdd(&accG[c0+1], G1);
  atomicAdd(&accG[c0+2], G2);  atomicAdd(&accG[c0+3], G3);
  atomicAdd(&accGm[c0+0], Gm0); atomicAdd(&accGm[c0+1], Gm1);
  atomicAdd(&accGm[c0+2], Gm2); atomicAdd(&accGm[c0+3], Gm3);
  float p1s = wave_sum32(p1), p2s = wave_sum32(p2);
  if (lane == 0) { atomicAdd(&accP[0], p1s); atomicAdd(&accP[1], p2s); }
  __syncthreads();

  // ---- channel-sum scalars (wave 0) ----
  if (tid < 32) {
    float sAWT = 0.f, sT = 0.f, sBG = 0.f, sBGm = 0.f;
#pragma unroll
    for (int k = 0; k < 4; ++k) {
      int c = tid * 4 + k;
      float t = accT[c];
      sAWT += s_aw[c] * t; sT += t;
      sBG  += s_bw[c] * accG[c];
      sBGm += s_bw[c] * accGm[c];
    }
    sAWT = wave_sum32(sAWT); sT = wave_sum32(sT);
    sBG  = wave_sum32(sBG);  sBGm = wave_sum32(sBGm);
    if (tid == 0) { s_scal[0]=sAWT; s_scal[1]=sT; s_scal[2]=sBG; s_scal[3]=sBGm; }
  }
  __syncthreads();

  // ---- scatter this block's partial contributions (global f32 atomics) ----
  if (act && tid < CDIM) {
    const int c = tid;
    if (bt != 3) {
      float wg;
      if (bt == 0)      wg = a * accT[c];
      else if (bt == 1) wg = a * s_aw[c] * (accT[c] - s_scal[0]);
      else              wg = a * bscale * accTm[c];
      atomicAdd(&out_gbw[off + c], wg);
    } else if (c < 3) {
      float ab = a * bscale;
      float gv = (c == 0) ? ab * s_scal[1] : (c == 1) ? ab * accP[0] : ab * accP[1];
      atomicAdd(&out_gbw[off + c], gv);
    }
    if (c == 0 && bt != 1) {
      float sg;
      if (bt == 0)      sg = a * s_scal[2];
      else if (bt == 2) sg = a * s_scal[3];
      else              sg = a * (w0 * s_scal[1] + w1 * accP[0] + w2 * accP[1]);
      atomicAdd(&out_gbs[bidc], sg);
    }
  }
}

extern "C" void kernel_launch(void* const* d_in, const int* in_sizes, int n_in,
                              void* d_out, int out_size, void* d_ws, size_t ws_size,
                              hipStream_t stream) {
  const float* g       = (const float*)d_in[0];
  const float* x       = (const float*)d_in[1];
  const float* alpha   = (const float*)d_in[2];
  const float* bwts    = (const float*)d_in[3];
  const float* bscales = (const float*)d_in[4];
  const int* lcs       = (const int*)d_in[5];
  const int* ids       = (const int*)d_in[6];
  const int* bts       = (const int*)d_in[7];
  const int* offsets   = (const int*)d_in[8];

  float* out_gi  = (float*)d_out;
  float* out_gbw = out_gi + (size_t)BDIM * HDIM;
  float* out_gbs = out_gbw + SDIM * CDIM;

  const int tail = SDIM * CDIM + SDIM;  // 8256 floats to zero (poisoned)
  kasmina_zero_tail<<<(tail + 255) / 256, 256, 0, stream>>>(out_gbw, tail);
  kasmina_bwd_main<<<SDIM * SPLITB, 256, 0, stream>>>(
      g, x, alpha, bwts, bscales, lcs, ids, bts, offsets,
      out_gi, out_gbw, out_gbs);
}